// STCA_ClassifyLoss_8993661518683
// MI455X (gfx1250) — compile-verified
//
#include <hip/hip_runtime.h>
#include <cstdint>

// ---------------------------------------------------------------------------
// STCA classification loss, CDNA5 (gfx1250).
// Streaming scan over vmem[B*N][T]; async global->LDS double-buffered pipeline.
// ---------------------------------------------------------------------------

#define THRESH   1.0f
#define CGAP     5
#define BT       128      // threads per block == rows per block
#define RPB      128      // rows per block
#define TS       64       // time-tile width (floats)
#define CPR      16       // b128 chunks per row per tile  (TS/4)
#define LSTRIDE  68       // LDS floats per row (padded; 68*4 % 16 == 0)
#define LOADS_PER_THREAD ((RPB * CPR) / BT)   // = 16 async b128 per thread/tile

// ---- gfx1250 async-tensor data movement (ISA ch.10.7 / CDNA5 ISA) ----
__device__ __forceinline__ void async_load_b128(unsigned lds_byte_off,
                                                const void* gaddr) {
    // GLOBAL_LOAD_ASYNC_TO_LDS_B128: per-lane 16B global -> 16B LDS, ASYNCcnt.
    asm volatile("global_load_async_to_lds_b128 %0, %1, off"
                 :
                 : "v"(lds_byte_off), "v"(gaddr)
                 : "memory");
}

template <int N>
__device__ __forceinline__ void wait_asynccnt() {
    asm volatile("s_wait_asynccnt %0" : : "n"(N) : "memory");
}

// Low 32 bits of a flat shared-aperture pointer == LDS byte offset (ISA 10.2).
__device__ __forceinline__ unsigned lds_off(const void* p) {
    return (unsigned)(uintptr_t)p;
}

// ---------------------------------------------------------------------------

__global__ void stca_zero_kernel(float* out, int n) {
    int i = blockIdx.x * blockDim.x + threadIdx.x;
    if (i < n) out[i] = 0.0f;
}

__device__ __forceinline__ void load_tile(const float* __restrict__ vmem,
                                          long long row0, long long rows,
                                          int T, int t0, float* bufp) {
    const int tid = threadIdx.x;
#pragma unroll
    for (int j = 0; j < LOADS_PER_THREAD; ++j) {
        int c  = j * BT + tid;       // chunk id within tile
        int r  = c >> 4;             // CPR = 16 chunks per row
        int c4 = c & 15;             // which 16B chunk in the row
        long long grow = row0 + r;
        if (grow >= rows) grow = rows - 1;   // clamp: keep EXEC full, addr valid
        const float* g = vmem + grow * (long long)T + (long long)(t0 + c4 * 4);
        unsigned l = lds_off(bufp + r * LSTRIDE + c4 * 4);
        async_load_b128(l, g);
    }
}

__global__ __launch_bounds__(BT)
void stca_loss_kernel(const float* __restrict__ vmem,
                      const int* __restrict__ labels,
                      float* __restrict__ out,
                      long long rows, int T) {
    __shared__ float buf[2][RPB * LSTRIDE];   // double-buffered tile (~68 KB)
    __shared__ float red[BT];

    const int tid = threadIdx.x;
    const long long row0 = (long long)blockIdx.x * RPB;
    const long long myrow = row0 + tid;
    const int nt = T / TS;

    // Prime the pipeline.
    load_tile(vmem, row0, rows, T, 0, buf[0]);

    // Per-row serial cluster state machine.
    float maxv  = -INFINITY;
    int   last_t = -(CGAP + 2);   // "no previous spike"
    int   cnt    = 0;             // spikes in current cluster
    float lastv  = 0.0f;          // v at current cluster's last spike
    int   best   = 0x7fffffff;    // smallest cluster size so far (strict <)
    float bestv  = 0.0f;          // v at that cluster's last spike
    bool  has    = false;         // any spike seen

    for (int k = 0; k < nt; ++k) {
        float* cur = buf[k & 1];
        if (k + 1 < nt) {
            // Kick tile k+1 into the other buffer (it was consumed at k-1 and
            // the trailing barrier of iteration k-1 ordered that for all waves),
            // then wait until only those 16 new per-wave ops are outstanding:
            // async loads complete in order, so tile k is resident.
            load_tile(vmem, row0, rows, T, (k + 1) * TS, buf[(k + 1) & 1]);
            wait_asynccnt<LOADS_PER_THREAD>();
        } else {
            wait_asynccnt<0>();
        }
        __syncthreads();

        const float* r = cur + tid * LSTRIDE;
        const int tbase = k * TS;
#pragma unroll 8
        for (int u = 0; u < TS; ++u) {
            float v = r[u];
            maxv = fmaxf(maxv, v);
            if (v >= THRESH) {
                int t = tbase + u;
                if (t - last_t > CGAP) {
                    // Previous cluster (if any) ended at last_t with `cnt`
                    // spikes; keep earliest-smallest (strict <  == first on tie,
                    // matching argmin over time-ordered cluster ends).
                    if (has && cnt < best) { best = cnt; bestv = lastv; }
                    cnt = 1;
                } else {
                    cnt++;
                }
                last_t = t;
                lastv  = v;
                has    = true;
            }
        }
        __syncthreads();   // everyone done with `cur` before it is refilled
    }
    // Finalize the trailing cluster.
    if (has && cnt < best) { best = cnt; bestv = lastv; }

    float c = 0.0f;
    if (myrow < rows) {
        int lab = labels[myrow];
        if (lab == 1 && !has)      c = THRESH - maxv;   // miss term
        else if (lab == 0 && has)  c = bestv - THRESH;  // false-positive term
    }

    // Block reduction -> one atomic per block.
    red[tid] = c;
    __syncthreads();
#pragma unroll
    for (int s = BT / 2; s > 0; s >>= 1) {
        if (tid < s) red[tid] += red[tid + s];
        __syncthreads();
    }
    if (tid == 0) atomicAdd(out, red[0]);
}

// ---------------------------------------------------------------------------

extern "C" void kernel_launch(void* const* d_in, const int* in_sizes, int n_in,
                              void* d_out, int out_size, void* d_ws, size_t ws_size,
                              hipStream_t stream) {
    const float* vmem   = (const float*)d_in[0];
    const int*   labels = (const int*)d_in[1];
    float*       out    = (float*)d_out;

    const long long rows = (long long)in_sizes[1];            // B*N = 32768
    const int T = (int)((long long)in_sizes[0] / rows);       // 1024

    // Zero the scalar accumulator (harness poisons d_out; atomics accumulate).
    stca_zero_kernel<<<(out_size + 63) / 64, 64, 0, stream>>>(out, out_size);

    const int nblocks = (int)((rows + RPB - 1) / RPB);        // 256 blocks
    stca_loss_kernel<<<nblocks, BT, 0, stream>>>(vmem, labels, out, rows, T);
}